// IndependentSemanticCodebooks_1125281431599
// MI455X (gfx1250) — compile-verified
//
#include <hip/hip_runtime.h>
#include <hip/hip_bf16.h>

typedef __attribute__((ext_vector_type(2))) float v2f;
typedef __attribute__((ext_vector_type(8))) float v8f;

#define NB 262144          // batch rows per group
#define DD 64              // feature dim
#define NG 10              // groups
#define ROWS_PER_BLOCK 128 // 8 waves * 16 rows

// Codebook sizes per group
__device__ __constant__ int c_K[NG] = {32, 32, 64, 128, 64, 128, 64, 64, 64, 64};
// (1 + cost) / (B * D) per group
__device__ __constant__ float c_scale[NG] = {
    1.5f / 16777216.0f, 1.5f / 16777216.0f,
    1.4f / 16777216.0f, 1.4f / 16777216.0f,
    1.4f / 16777216.0f, 1.4f / 16777216.0f,
    1.8f / 16777216.0f, 1.8f / 16777216.0f,
    1.8f / 16777216.0f, 1.8f / 16777216.0f};

struct VQParams {
    const float* x[NG];
    const float* w[NG];
    float* q;     // (10, B, 64) quantized
    float* loss;  // scalar
    float* idx;   // (10, B) indices as float
};

__global__ void vq_zero_loss(float* loss) {
    if (threadIdx.x == 0 && blockIdx.x == 0) *loss = 0.0f;
}

extern "C" __global__ __launch_bounds__(256)
void vq_wmma_kernel(VQParams p) {
    __shared__ float s_w[128 * DD];   // codebook, up to 128 x 64 fp32 = 32 KB
    __shared__ float s_wsq[128];      // |w_k|^2
    __shared__ float s_loss;

    const int g   = blockIdx.y;
    const int K   = c_K[g];
    const int tid = threadIdx.x;

    const float* __restrict__ xg = p.x[g];
    const float* __restrict__ wg = p.w[g];

    if (tid == 0) s_loss = 0.0f;
    // Stage codebook into LDS
    for (int i = tid; i < K * DD; i += 256) s_w[i] = wg[i];
    __syncthreads();
    // Precompute squared norms
    if (tid < K) {
        float s = 0.0f;
        #pragma unroll 8
        for (int d = 0; d < DD; ++d) { float v = s_w[tid * DD + d]; s += v * v; }
        s_wsq[tid] = s;
    }
    __syncthreads();

    const int lane = tid & 31;
    const int wave = tid >> 5;
    const int L    = lane & 15;   // row-within-tile / N index
    const int hi   = lane >> 4;   // lane-half selector
    const int row  = blockIdx.x * ROWS_PER_BLOCK + wave * 16 + L;

    // Load this lane's slice of the 16x64 x tile in WMMA B-operand layout:
    // lane (L,hi) holds x[row][4j + 2*hi + {0,1}] for j = 0..15.
    v2f xf[16];
    const float* xrow = xg + (size_t)row * DD + 2 * hi;
    #pragma unroll
    for (int j = 0; j < 16; ++j)
        xf[j] = *(const v2f*)(xrow + 4 * j);

    // Running argmax of score = 2*dot(w_k,x) - |w_k|^2 (== argmin distance)
    float best  = -3.402823466e38f;
    int   bestk = 0;

    for (int kk = 0; kk < K; kk += 16) {
        v8f c = {0.f, 0.f, 0.f, 0.f, 0.f, 0.f, 0.f, 0.f};
        const float* wl = &s_w[(size_t)(kk + L) * DD + 2 * hi];
        #pragma unroll
        for (int j = 0; j < 16; ++j) {
            // A-fragment (16x4 f32): lane (L,hi) holds w[kk+L][4j + 2*hi + {0,1}]
            v2f a = *(const v2f*)(wl + 4 * j);
            // D = A(16x4) x B(4x16) + C  -> C[m][n] = dot(w[kk+m], x[row n])
            c = __builtin_amdgcn_wmma_f32_16x16x4_f32(
                    false, a, false, xf[j], (short)0, c, false, false);
        }
        // C layout: lane n (lo): vgpr r = M=r ; lane n+16: vgpr r = M=r+8
        #pragma unroll
        for (int r = 0; r < 8; ++r) {
            int k = kk + r + 8 * hi;
            float s = 2.0f * c[r] - s_wsq[k];
            if (s > best) { best = s; bestk = k; }  // ascending k -> first-min tie rule
        }
    }

    // Merge lane pair (L covers smaller m's is not guaranteed across halves -> tie on index)
    {
        float ob = __shfl_xor(best, 16);
        int   ok = __shfl_xor(bestk, 16);
        if (ob > best || (ob == best && ok < bestk)) { best = ob; bestk = ok; }
    }

    // Loss partial from register-resident x (no x re-read):
    // lanes (L,0) and (L,16) jointly cover all 64 dims of row L.
    float lsum = 0.0f;
    {
        const float* wq = &s_w[(size_t)bestk * DD + 2 * hi];
        #pragma unroll
        for (int j = 0; j < 16; ++j) {
            v2f wv = *(const v2f*)(wq + 4 * j);
            float d0 = wv.x - xf[j].x;
            float d1 = wv.y - xf[j].y;
            lsum += d0 * d0 + d1 * d1;
        }
    }

    // Write quantized rows: broadcast winner per row, full wave streams the
    // 256B row from LDS -> perfectly coalesced global stores.
    {
        size_t qbase = ((size_t)g * NB + (size_t)blockIdx.x * ROWS_PER_BLOCK
                        + (size_t)wave * 16) * DD;
        for (int r = 0; r < 16; ++r) {
            int k = __shfl(bestk, r);
            v2f wv = *(const v2f*)(&s_w[(size_t)k * DD + 2 * lane]);
            *(v2f*)(p.q + qbase + (size_t)r * DD + 2 * lane) = wv;
        }
        if (hi == 0)
            p.idx[(size_t)g * NB + row] = (float)bestk;
    }

    // Wave shfl-tree reduce -> LDS atomic -> one global atomic per block
    #pragma unroll
    for (int off = 16; off >= 1; off >>= 1)
        lsum += __shfl_xor(lsum, off);
    if (lane == 0) atomicAdd(&s_loss, lsum);
    __syncthreads();
    if (tid == 0) atomicAdd(p.loss, s_loss * c_scale[g]);
}

extern "C" void kernel_launch(void* const* d_in, const int* in_sizes, int n_in,
                              void* d_out, int out_size, void* d_ws, size_t ws_size,
                              hipStream_t stream) {
    (void)in_sizes; (void)n_in; (void)d_ws; (void)ws_size; (void)out_size;

    VQParams p;
    for (int i = 0; i < NG; ++i) {
        p.x[i] = (const float*)d_in[i];        // x_* in group order
        p.w[i] = (const float*)d_in[NG + i];   // w_* in group order
    }
    float* out = (float*)d_out;
    const size_t qoff = (size_t)NG * NB * DD;  // 167,772,160 floats
    p.q    = out;
    p.loss = out + qoff;
    p.idx  = out + qoff + 1;

    vq_zero_loss<<<1, 64, 0, stream>>>(p.loss);

    dim3 grid(NB / ROWS_PER_BLOCK, NG);  // (2048, 10)
    vq_wmma_kernel<<<grid, 256, 0, stream>>>(p);
}